// DualEEGTransformer_50869592655007
// MI455X (gfx1250) — compile-verified
//
#include <hip/hip_runtime.h>
#include <math.h>

// ---------------------------------------------------------------------------
// EEG connectivity features on MI455X (gfx1250), built around f32 WMMA GEMMs.
//   Stage 1: rfft bins 4..363 of all 256 channels via DFT-GEMM (WMMA 16x16x4)
//   Stage 2 (per band): analytic signal via inverse-DFT GEMM (WMMA)
//   Stage 3: per-channel prep (phase, cos/sin, power, z-norms) + reductions
//   Stage 4: plv/pcorr/tcorr as C x C Gram matrices (WMMA, 5 accumulators)
//   Stage 5: pli/wpli/pdiff/coh pairwise reductions (one wave per pair)
// ---------------------------------------------------------------------------

typedef float v2f __attribute__((ext_vector_type(2)));
typedef float v8f __attribute__((ext_vector_type(8)));

#define T_LEN 2048
#define NCH2  256          // both signals: 2 * B(4) * C(32)
#define XCOLS 720          // 360 bins * (re,im); bin k = 4 + col/2
#define EPS   1e-8f
#define TWO_PI 6.2831853071795864769f

static __device__ __forceinline__ v8f wmma4(v2f a, v2f b, v8f c) {
  // D = A(16x4) * B(4x16) + C, fp32 throughout.
  return __builtin_amdgcn_wmma_f32_16x16x4_f32(false, a, false, b, (short)0, c,
                                               false, false);
}

// ---- forward DFT weights, stored transposed: Wt[col][t], col = 2*bin(+im) --
__global__ void k_wgen(float* __restrict__ Wt) {
  int idx = blockIdx.x * blockDim.x + threadIdx.x;
  if (idx >= XCOLS * T_LEN) return;
  int col = idx / T_LEN, t = idx % T_LEN;
  int k = 4 + (col >> 1);
  int ai = (k * t) & (T_LEN - 1);              // exact angle reduction
  float ang = TWO_PI * (float)ai / (float)T_LEN;
  float s, c;
  __sincosf(ang, &s, &c);
  Wt[idx] = (col & 1) ? -s : c;                // X = sum x*(cos - i sin)
}

// ---- DFT as GEMM: X[ch][col] = sum_t E[ch][t] * Wt[col][t] ----------------
__global__ void k_dft(const float* __restrict__ e1, const float* __restrict__ e2,
                      const float* __restrict__ Wt, float* __restrict__ X) {
  int mt = blockIdx.x;            // 16 channel tiles (0..7 sig1, 8..15 sig2)
  int nt = blockIdx.y;            // 45 column tiles
  int l = threadIdx.x, lr = l & 15, lh = l >> 4;
  const float* sig = (mt < 8) ? e1 : e2;
  const float* arow = sig + (size_t)((mt & 7) * 16 + lr) * T_LEN + 2 * lh;
  const float* brow = Wt + (size_t)(nt * 16 + lr) * T_LEN + 2 * lh;
  v8f acc = {};
  for (int kb = 0; kb < T_LEN; kb += 4) {
    v2f a = *(const v2f*)(arow + kb);          // A[M=lr][K=kb+2*lh+v]
    v2f b = *(const v2f*)(brow + kb);          // B[K][N=lr] from Wt (transposed)
    acc = wmma4(a, b, acc);
  }
  for (int r = 0; r < 8; ++r) {
    int ch = mt * 16 + r + 8 * lh;
    X[(size_t)ch * XCOLS + nt * 16 + lr] = acc[r];
  }
}

// ---- per-band inverse-DFT weights, stored [t][kk]; kk band-relative -------
__global__ void k_bgen(float* __restrict__ Bar, float* __restrict__ Bai,
                       int k0, int bins) {
  int idx = blockIdx.x * blockDim.x + threadIdx.x;
  if (idx >= T_LEN * XCOLS) return;
  int t = idx / XCOLS, kk = idx % XCOLS;
  int m = kk >> 1;
  float vr = 0.f, vi = 0.f;
  if (m < bins) {
    int k = k0 + m;
    int ai = (k * t) & (T_LEN - 1);
    float ang = TWO_PI * (float)ai / (float)T_LEN;
    float s, c;
    __sincosf(ang, &s, &c);
    const float sc = 2.0f / (float)T_LEN;      // analytic scale 2/T
    if (kk & 1) { vr = -sc * s; vi = sc * c; } // Xi columns
    else        { vr =  sc * c; vi = sc * s; } // Xr columns
  }
  Bar[idx] = vr;
  Bai[idx] = vi;
}

// ---- analytic signal GEMM: a[ch][t] = sum_kk X[ch][aoff+kk] * B[kk][t] ----
__global__ void k_analytic(const float* __restrict__ X, const float* __restrict__ Bar,
                           const float* __restrict__ Bai, float* __restrict__ ar,
                           float* __restrict__ ai, int aoff, int kpad) {
  int mt = blockIdx.x;            // 16 channel tiles
  int nt = blockIdx.y;            // 128 time tiles
  int l = threadIdx.x, lr = l & 15, lh = l >> 4;
  const float* arow = X + (size_t)(mt * 16 + lr) * XCOLS + aoff + 2 * lh;
  const float* brrow = Bar + (size_t)(nt * 16 + lr) * XCOLS + 2 * lh;
  const float* birow = Bai + (size_t)(nt * 16 + lr) * XCOLS + 2 * lh;
  v8f accR = {}, accI = {};
  for (int kb = 0; kb < kpad; kb += 4) {
    v2f a  = *(const v2f*)(arow + kb);
    v2f br = *(const v2f*)(brrow + kb);
    v2f bi = *(const v2f*)(birow + kb);
    accR = wmma4(a, br, accR);
    accI = wmma4(a, bi, accI);
  }
  for (int r = 0; r < 8; ++r) {
    int ch = mt * 16 + r + 8 * lh;
    int t = nt * 16 + lr;
    ar[(size_t)ch * T_LEN + t] = accR[r];
    ai[(size_t)ch * T_LEN + t] = accI[r];
  }
}

// ---- per-channel derived quantities + mean/std reductions -----------------
__global__ void k_prep(const float* __restrict__ ar, const float* __restrict__ ai,
                       float* __restrict__ ph, float* __restrict__ cc,
                       float* __restrict__ ss, float* __restrict__ pw,
                       float* __restrict__ zpw, float* __restrict__ zb,
                       float* __restrict__ spw) {
  int ch = blockIdx.x;
  int tid = threadIdx.x;                       // 256 threads, 8 elems each
  __shared__ float red[256];
  __shared__ float tot[4];
  size_t base = (size_t)ch * T_LEN;
  float s1 = 0.f, s2 = 0.f, s3 = 0.f, s4 = 0.f;
  for (int e = 0; e < 8; ++e) {
    int t = tid + e * 256;
    float x = ar[base + t], y = ai[base + t];
    float p = x * x + y * y;                   // instantaneous power
    float inv = rsqrtf(p + 1e-30f);
    ph[base + t] = atan2f(y, x);
    cc[base + t] = x * inv;                    // cos(phase)
    ss[base + t] = y * inv;                    // sin(phase)
    pw[base + t] = p;
    s1 += p; s2 += p * p; s3 += x; s4 += x * x;
  }
  float sums[4] = {s1, s2, s3, s4};
  for (int q = 0; q < 4; ++q) {
    red[tid] = sums[q];
    __syncthreads();
    for (int off = 128; off > 0; off >>= 1) {
      if (tid < off) red[tid] += red[tid + off];
      __syncthreads();
    }
    if (tid == 0) tot[q] = red[0];
    __syncthreads();
  }
  const float invT = 1.0f / (float)T_LEN;
  float mpw = tot[0] * invT;
  float vpw = (tot[1] - tot[0] * tot[0] * invT) / (float)(T_LEN - 1);
  float zspw = 1.0f / (sqrtf(fmaxf(vpw, 0.f)) + EPS);       // ddof=1 std
  float mb = tot[2] * invT;
  float vb = (tot[3] - tot[2] * tot[2] * invT) / (float)(T_LEN - 1);
  float zsb = 1.0f / (sqrtf(fmaxf(vb, 0.f)) + EPS);
  for (int e = 0; e < 8; ++e) {
    int t = tid + e * 256;
    zpw[base + t] = (pw[base + t] - mpw) * zspw;
    zb[base + t] = (ar[base + t] - mb) * zsb;
  }
  if (tid == 0) spw[ch] = tot[0];
}

// ---- plv / pcorr / tcorr: C x C Gram matrices via WMMA --------------------
__global__ void k_pairmm(const float* __restrict__ cc, const float* __restrict__ ss,
                         const float* __restrict__ zpw, const float* __restrict__ zb,
                         float* __restrict__ out, int band) {
  int blk = blockIdx.x;                        // b*4 + ti*2 + tj
  int b = blk >> 2, ti = (blk >> 1) & 1, tj = blk & 1;
  int l = threadIdx.x, lr = l & 15, lh = l >> 4;
  size_t rowA = (size_t)(b * 32 + ti * 16 + lr) * T_LEN + 2 * lh;        // sig1
  size_t rowB = (size_t)(128 + b * 32 + tj * 16 + lr) * T_LEN + 2 * lh;  // sig2
  v8f aRe = {}, aImA = {}, aImB = {}, aP = {}, aT = {};
  for (int kb = 0; kb < T_LEN; kb += 4) {
    v2f ac = *(const v2f*)(cc + rowA + kb);
    v2f as = *(const v2f*)(ss + rowA + kb);
    v2f ap = *(const v2f*)(zpw + rowA + kb);
    v2f ab = *(const v2f*)(zb + rowA + kb);
    v2f bc = *(const v2f*)(cc + rowB + kb);
    v2f bs = *(const v2f*)(ss + rowB + kb);
    v2f bp = *(const v2f*)(zpw + rowB + kb);
    v2f bb = *(const v2f*)(zb + rowB + kb);
    aRe = wmma4(ac, bc, aRe);                  // re  = c1 c2^T + s1 s2^T
    aRe = wmma4(as, bs, aRe);
    aImA = wmma4(as, bc, aImA);                // im  = s1 c2^T - c1 s2^T
    aImB = wmma4(ac, bs, aImB);                //  (split: f32 WMMA has no A-neg)
    aP = wmma4(ap, bp, aP);                    // pcorr
    aT = wmma4(ab, bb, aT);                    // tcorr
  }
  const float invT = 1.0f / (float)T_LEN;
  size_t ob = ((size_t)b * 6 + band) * 7 * 1024;
  for (int r = 0; r < 8; ++r) {
    int i = ti * 16 + r + 8 * lh;
    int j = tj * 16 + lr;
    float re = aRe[r], im = aImA[r] - aImB[r];
    out[ob + 0 * 1024 + i * 32 + j] = sqrtf(re * re + im * im) * invT;  // plv
    out[ob + 4 * 1024 + i * 32 + j] = aP[r] * invT;                     // pcorr
    out[ob + 6 * 1024 + i * 32 + j] = aT[r] * invT;                     // tcorr
  }
}

// ---- pli / wpli / pdiff / coh: one wave per (b,i,j) pair ------------------
__global__ void k_pairred(const float* __restrict__ ph, const float* __restrict__ pw,
                          const float* __restrict__ X, const float* __restrict__ spw,
                          float* __restrict__ out, int band, int b0off, int bins) {
  int blk = blockIdx.x;                        // 4096 = 4*32*32
  int b = blk >> 10, i = (blk >> 5) & 31, j = blk & 31;
  int l = threadIdx.x;
  size_t r1 = (size_t)(b * 32 + i) * T_LEN;
  size_t r2 = (size_t)(128 + b * 32 + j) * T_LEN;
  float aP = 0.f, aW = 0.f, aD = 0.f, aC = 0.f;
  for (int t = l; t < T_LEN; t += 32) {
    float d = ph[r1 + t] - ph[r2 + t];
    float sg = (d > 0.f) ? 1.f : ((d < 0.f) ? -1.f : 0.f);
    aP += sg;
    aW += sg * (pw[r1 + t] + pw[r2 + t]);
    aD += fabsf(d);
  }
  size_t x1 = (size_t)(b * 32 + i) * XCOLS + 2 * b0off;
  size_t x2 = (size_t)(128 + b * 32 + j) * XCOLS + 2 * b0off;
  for (int m = l; m < bins; m += 32) {
    float xr = X[x1 + 2 * m], xi = X[x1 + 2 * m + 1];
    float yr = X[x2 + 2 * m], yi = X[x2 + 2 * m + 1];
    float pp = (xr * xr + xi * xi) * (yr * yr + yi * yi);
    aC += pp / (pp + EPS);                     // out-of-band bins contribute 0
  }
  for (int off = 16; off > 0; off >>= 1) {
    aP += __shfl_xor(aP, off, 32);
    aW += __shfl_xor(aW, off, 32);
    aD += __shfl_xor(aD, off, 32);
    aC += __shfl_xor(aC, off, 32);
  }
  if (l == 0) {
    size_t ob = ((size_t)b * 6 + band) * 7 * 1024 + (size_t)i * 32 + j;
    out[ob + 1 * 1024] = fabsf(aP) / (float)T_LEN;                       // pli
    out[ob + 2 * 1024] = fabsf(0.5f * aW) /
        (0.5f * (spw[b * 32 + i] + spw[128 + b * 32 + j]) + EPS);        // wpli
    out[ob + 3 * 1024] = aC / 1025.0f;                                   // coh
    out[ob + 5 * 1024] = aD / (float)T_LEN;                              // pdiff
  }
}

extern "C" void kernel_launch(void* const* d_in, const int* in_sizes, int n_in,
                              void* d_out, int out_size, void* d_ws, size_t ws_size,
                              hipStream_t stream) {
  (void)in_sizes; (void)n_in; (void)out_size; (void)ws_size;
  const float* e1 = (const float*)d_in[0];
  const float* e2 = (const float*)d_in[1];
  float* out = (float*)d_out;

  float* p = (float*)d_ws;
  float* Wt  = p; p += (size_t)XCOLS * T_LEN;      // 720 x 2048 (transposed)
  float* X   = p; p += (size_t)NCH2 * XCOLS;       // 256 x 720 spectrum
  float* Bar = p; p += (size_t)T_LEN * XCOLS;      // band weights (re path)
  float* Bai = p; p += (size_t)T_LEN * XCOLS;      // band weights (im path)
  float* ar  = p; p += (size_t)NCH2 * T_LEN;
  float* ai  = p; p += (size_t)NCH2 * T_LEN;
  float* ph  = p; p += (size_t)NCH2 * T_LEN;
  float* cc  = p; p += (size_t)NCH2 * T_LEN;
  float* ssb = p; p += (size_t)NCH2 * T_LEN;
  float* pw  = p; p += (size_t)NCH2 * T_LEN;
  float* zpw = p; p += (size_t)NCH2 * T_LEN;
  float* zb  = p; p += (size_t)NCH2 * T_LEN;
  float* spw = p; p += 256;

  k_wgen<<<(XCOLS * T_LEN + 255) / 256, 256, 0, stream>>>(Wt);
  k_dft<<<dim3(16, 45), 32, 0, stream>>>(e1, e2, Wt, X);

  // bands: (0.5,45) (0.5,4) (4,8) (8,13) (13,30) (30,45); bin = freq/0.125
  static const int bk0[6]  = {4, 4, 32, 64, 104, 240};
  static const int bbin[6] = {357, 29, 33, 41, 137, 121};
  for (int band = 0; band < 6; ++band) {
    int k0 = bk0[band], bins = bbin[band];
    int kpad = (2 * bins + 3) & ~3;              // K padded to WMMA K=4 steps
    int aoff = 2 * (k0 - 4);                     // band start col in X
    k_bgen<<<(T_LEN * XCOLS + 255) / 256, 256, 0, stream>>>(Bar, Bai, k0, bins);
    k_analytic<<<dim3(16, 128), 32, 0, stream>>>(X, Bar, Bai, ar, ai, aoff, kpad);
    k_prep<<<256, 256, 0, stream>>>(ar, ai, ph, cc, ssb, pw, zpw, zb, spw);
    k_pairmm<<<16, 32, 0, stream>>>(cc, ssb, zpw, zb, out, band);
    k_pairred<<<4096, 32, 0, stream>>>(ph, pw, X, spw, out, band, k0 - 4, bins);
  }
}